// ContrastiveLoss_50955491999905
// MI455X (gfx1250) — compile-verified
//
#include <hip/hip_runtime.h>

typedef float v2f __attribute__((ext_vector_type(2)));
typedef float v8f __attribute__((ext_vector_type(8)));

static constexpr int   BSZ    = 1024;   // batch
static constexpr int   CH     = 256;    // channels
static constexpr float EPS    = 1e-6f;
static constexpr float MARGIN = 1.0f;

// ---------------------------------------------------------------------------
// Kernel 1: per-row sum and squared-norm.  One block per row, 256 threads.
// ---------------------------------------------------------------------------
__global__ __launch_bounds__(256)
void cl_row_stats(const float* __restrict__ X,
                  float* __restrict__ norms,
                  float* __restrict__ sums) {
    __shared__ float s1[256];
    __shared__ float s2[256];
    const int row = blockIdx.x;
    const int t   = threadIdx.x;
    const float x = X[row * CH + t];
    s1[t] = x;
    s2[t] = x * x;
    __syncthreads();
    for (int off = 128; off > 0; off >>= 1) {
        if (t < off) { s1[t] += s1[t + off]; s2[t] += s2[t + off]; }
        __syncthreads();
    }
    if (t == 0) { sums[row] = s1[0]; norms[row] = s2[0]; }
}

// ---------------------------------------------------------------------------
// Kernel 2: Gram tiles via V_WMMA_F32_16X16X4_F32 + fused loss reduction.
// Block = 256 threads = 8 waves; each wave computes one 16x16 tile of
// G = X * X^T.  Block covers a 32 (i) x 64 (j) region -> grid (32, 16).
//
// f32 WMMA fragment layout (ISA 7.12.2):
//   A 16x4 : lane L holds A[M = L&15][K = 2*(L>>4) + {0,1}] in v0,v1
//   B 4x16 : lane L holds B[K = 2*(L>>4) + {0,1}][N = L&15] in v0,v1
// With B[k][n] = X[j0+n][k], both fragments load identically:
//   float2 from row (base + (L&15)), cols k + 2*(L>>4).
// D 16x16 : acc[r] at lane L is (i = i0 + r + 8*(L>=16), j = j0 + (L&15)).
// ---------------------------------------------------------------------------
__global__ __launch_bounds__(256)
void cl_gram_loss(const float* __restrict__ X,
                  const int*   __restrict__ labels,
                  const float* __restrict__ norms,
                  const float* __restrict__ sums,
                  float* __restrict__ partials) {
    const int lane = threadIdx.x & 31;
    const int wave = threadIdx.x >> 5;                 // 0..7
    const int tileI = blockIdx.x * 2 + (wave >> 2);    // 0..63
    const int tileJ = blockIdx.y * 4 + (wave & 3);     // 0..63
    const int i0 = tileI * 16;
    const int j0 = tileJ * 16;

    const int m  = lane & 15;          // row within fragment
    const int kh = (lane >> 4) << 1;   // 0 or 2: K sub-offset for this half-wave

    const float* __restrict__ arow = X + (size_t)(i0 + m) * CH + kh;
    const float* __restrict__ brow = X + (size_t)(j0 + m) * CH + kh;

    v8f acc = {};
#pragma unroll 8
    for (int k = 0; k < CH; k += 4) {
        v2f a = *(const v2f*)(arow + k);
        v2f b = *(const v2f*)(brow + k);
        // D = A x B + C   (16x16x4, f32)
        acc = __builtin_amdgcn_wmma_f32_16x16x4_f32(
            /*neg_a=*/false, a, /*neg_b=*/false, b,
            /*c_mod=*/(short)0, acc, /*reuse_a=*/false, /*reuse_b=*/false);
    }

    // ---- per-lane loss contributions from the 16x16 tile ----
    const int   j  = j0 + m;
    const float nj = norms[j];
    const float sj = sums[j];
    const int   lj = labels[j];
    const float epsConst = (float)CH * EPS * EPS;

    float part = 0.0f;
#pragma unroll
    for (int r = 0; r < 8; ++r) {
        const int   i  = i0 + r + ((lane >> 4) << 3);
        const float g  = acc[r];
        float d2 = norms[i] + nj - 2.0f * g
                 + 2.0f * EPS * (sj - sums[i]) + epsConst;
        d2 = fmaxf(d2, 0.0f);
        const float md = fmaxf(MARGIN - sqrtf(d2), 0.0f);
        const float contrib = (labels[i] == lj) ? d2 : md * md;
        part += (i != j) ? contrib : 0.0f;
    }

    // ---- wave32 reduction (fixed order -> deterministic) ----
#pragma unroll
    for (int off = 16; off > 0; off >>= 1)
        part += __shfl_down(part, off, 32);

    __shared__ float wsum[8];
    if (lane == 0) wsum[wave] = part;
    __syncthreads();
    if (threadIdx.x == 0) {
        float s = 0.0f;
#pragma unroll
        for (int w = 0; w < 8; ++w) s += wsum[w];
        partials[blockIdx.y * gridDim.x + blockIdx.x] = s;
    }
}

// ---------------------------------------------------------------------------
// Kernel 3: deterministic final reduction of block partials.
// ---------------------------------------------------------------------------
__global__ __launch_bounds__(256)
void cl_finalize(const float* __restrict__ partials, int n,
                 float* __restrict__ out) {
    __shared__ float s[256];
    float acc = 0.0f;
    for (int i = threadIdx.x; i < n; i += 256) acc += partials[i];
    s[threadIdx.x] = acc;
    __syncthreads();
    for (int off = 128; off > 0; off >>= 1) {
        if (threadIdx.x < off) s[threadIdx.x] += s[threadIdx.x + off];
        __syncthreads();
    }
    if (threadIdx.x == 0)
        out[0] = s[0] / (float)((long)BSZ * (BSZ - 1));
}

// ---------------------------------------------------------------------------
extern "C" void kernel_launch(void* const* d_in, const int* in_sizes, int n_in,
                              void* d_out, int out_size, void* d_ws, size_t ws_size,
                              hipStream_t stream) {
    const float* X      = (const float*)d_in[0];
    const int*   labels = (const int*)d_in[1];

    float* ws       = (float*)d_ws;
    float* norms    = ws;              // 1024 floats
    float* sums     = ws + BSZ;        // 1024 floats
    float* partials = ws + 2 * BSZ;    // 512 floats

    cl_row_stats<<<BSZ, 256, 0, stream>>>(X, norms, sums);

    dim3 grid(BSZ / 32, BSZ / 64);     // (32, 16) = 512 blocks
    cl_gram_loss<<<grid, 256, 0, stream>>>(X, labels, norms, sums, partials);

    cl_finalize<<<1, 256, 0, stream>>>(partials,
                                       (BSZ / 32) * (BSZ / 64),
                                       (float*)d_out);
}